// PairformerLayer_56736517980632
// MI455X (gfx1250) — compile-verified
//
#include <hip/hip_runtime.h>
#include <math.h>

typedef float v2f __attribute__((ext_vector_type(2)));
typedef float v8f __attribute__((ext_vector_type(8)));

#define N_  256
#define CZ_ 128
#define CS_ 384
#define ZS_ (N_ * N_)
#define EPS_ 1e-5f

// ---------------------------------------------------------------------------
// WMMA helpers: V_WMMA_F32_16X16X4_F32  (fp32 A 16x4, B 4x16, C/D 16x16 f32)
// A layout: lane<16 -> M=lane, K={k0,k0+1}; lane>=16 -> M=lane-16, K={k0+2,k0+3}
// B layout: lane<16 -> K={k0,k0+1}, N=lane; lane>=16 -> K={k0+2,k0+3}, N=lane-16
// C layout: c[r] -> M = r + 8*(lane>=16), N = lane%16
// ---------------------------------------------------------------------------
__device__ __forceinline__ v8f wmma4(v2f a, v2f b, v8f c) {
  return __builtin_amdgcn_wmma_f32_16x16x4_f32(false, a, false, b, (short)0, c,
                                               false, false);
}

__device__ __forceinline__ v2f load_a_tile(const float* A, int lda, int row0, int k0) {
  int lane = threadIdx.x & 31;
  int m = lane & 15;
  int kk = k0 + ((lane >> 4) << 1);
  const float* p = A + (size_t)(row0 + m) * lda + kk;
  v2f r; r.x = p[0]; r.y = p[1];
  return r;
}

__device__ __forceinline__ v2f load_b_tile(const float* B, int ldb, int k0, int col0) {
  int lane = threadIdx.x & 31;
  int n = lane & 15;
  int kk = k0 + ((lane >> 4) << 1);
  const float* p = B + (size_t)kk * ldb + col0 + n;
  v2f r; r.x = p[0]; r.y = p[ldb];
  return r;
}

__device__ __forceinline__ float sigmoidf_(float x) {
  return 1.0f / (1.0f + __expf(-x));
}

// ---------------------------------------------------------------------------
// Generic GEMM: C(M,Nn) = epilogue(A(M,K) @ B(K,Nn))
// Wave tile 16x64: one A fragment feeds 4 WMMAs (A register reuse).
// ---------------------------------------------------------------------------
enum { EPI_STORE = 0, EPI_ACC = 1, EPI_SIG = 2, EPI_BIAS = 3 };

template <int EPI, bool TRANS_OUT>
__global__ void gemm_kernel(const float* __restrict__ A, const float* __restrict__ B,
                            const float* __restrict__ bias, float* __restrict__ C,
                            int K, int lda, int ldb, int ldc) {
  int row0 = (blockIdx.y * blockDim.y + threadIdx.y) * 16;
  int col0 = blockIdx.x * 64;
  int lane = threadIdx.x & 31;
  int m = lane & 15;
  v8f acc[4] = {};
  const float* aprefetch = A + (size_t)(row0 + m) * lda + ((lane >> 4) << 1);
  for (int k0 = 0; k0 < K; k0 += 4) {
    v2f a = load_a_tile(A, lda, row0, k0);
    if (k0 + 32 < K) __builtin_prefetch(aprefetch + k0 + 32, 0, 0);  // global_prefetch_b8
#pragma unroll
    for (int c = 0; c < 4; ++c)
      acc[c] = wmma4(a, load_b_tile(B, ldb, k0, col0 + 16 * c), acc[c]);
  }
  int n = lane & 15, half = lane >> 4;
#pragma unroll
  for (int c = 0; c < 4; ++c) {
#pragma unroll
    for (int r = 0; r < 8; ++r) {
      int lr = row0 + half * 8 + r;
      int col = col0 + 16 * c + n;
      float v = acc[c][r];
      if (EPI == EPI_BIAS) v += bias[col];
      if (EPI == EPI_SIG) v = sigmoidf_(v);
      if (TRANS_OUT) {
        int tr = ((lr & 255) << 8) | (lr >> 8);  // (i*N+q) -> (q*N+i), N=256
        C[(size_t)tr * ldc + col] += v;
      } else if (EPI == EPI_ACC) {
        C[(size_t)lr * ldc + col] += v;
      } else {
        C[(size_t)lr * ldc + col] = v;
      }
    }
  }
}

// ---------------------------------------------------------------------------
// Dual GEMM with gating:  GM_SIG:  out = (A1@B1) * sigmoid(A2@B2)
//                         GM_SILU: out = silu(A1@B1) * (A2@B2)
// Wave tile 16x32 (2 col tiles x 2 products). SAME_A: A1==A2, load A once.
// ---------------------------------------------------------------------------
enum { GM_SIG = 0, GM_SILU = 1 };

template <int GM, bool ACC, bool SAME_A>
__global__ void gemm_dual_kernel(const float* __restrict__ A1, const float* __restrict__ B1,
                                 const float* __restrict__ A2, const float* __restrict__ B2,
                                 float* __restrict__ C, int K, int lda, int ldb, int ldc) {
  int row0 = (blockIdx.y * blockDim.y + threadIdx.y) * 16;
  int col0 = blockIdx.x * 32;
  v8f p[2] = {}, g[2] = {};
  for (int k0 = 0; k0 < K; k0 += 4) {
    v2f a1 = load_a_tile(A1, lda, row0, k0);
    v2f a2 = SAME_A ? a1 : load_a_tile(A2, lda, row0, k0);
#pragma unroll
    for (int c = 0; c < 2; ++c) {
      p[c] = wmma4(a1, load_b_tile(B1, ldb, k0, col0 + 16 * c), p[c]);
      g[c] = wmma4(a2, load_b_tile(B2, ldb, k0, col0 + 16 * c), g[c]);
    }
  }
  int lane = threadIdx.x & 31;
  int n = lane & 15, half = lane >> 4;
#pragma unroll
  for (int c = 0; c < 2; ++c) {
#pragma unroll
    for (int r = 0; r < 8; ++r) {
      int lr = row0 + half * 8 + r;
      float pv = p[c][r], gv = g[c][r];
      float o = (GM == GM_SIG) ? pv * sigmoidf_(gv) : pv * sigmoidf_(pv) * gv;
      float* cp = C + (size_t)lr * ldc + col0 + 16 * c + n;
      if (ACC) *cp += o; else *cp = o;
    }
  }
}

// ---------------------------------------------------------------------------
// Triangle einsum:  !ENDING: O[i,j,d] = sum_k a[i,k,d]*b[j,k,d]
//                    ENDING: O[i,j,d] = sum_k a[k,i,d]*b[k,j,d]
// a = AB[:, 0:128], b = AB[:, 128:256], ld(AB)=256.
// Wave tile 16(i) x 64(j) at fixed d: A gather reused across 4 j-tiles.
// ---------------------------------------------------------------------------
template <bool ENDING>
__global__ void trimul_einsum_kernel(const float* __restrict__ AB, float* __restrict__ O) {
  int d  = blockIdx.z;
  int i0 = blockIdx.y * 16;
  int j0 = threadIdx.y * 64;   // 4 waves cover all 256 j
  int lane = threadIdx.x & 31;
  int mn = lane & 15, half = lane >> 4;
  const int LD = 2 * CZ_;
  v8f acc[4] = {};
  for (int k0 = 0; k0 < N_; k0 += 4) {
    int ka = k0 + half * 2;
    v2f a;
    if (!ENDING) {
      const float* pa = AB + ((size_t)(i0 + mn) * N_ + ka) * LD + d;
      a.x = pa[0]; a.y = pa[LD];
    } else {
      const float* pa = AB + ((size_t)ka * N_ + i0 + mn) * LD + d;
      a.x = pa[0]; a.y = pa[(size_t)N_ * LD];
    }
#pragma unroll
    for (int c = 0; c < 4; ++c) {
      v2f b;
      if (!ENDING) {
        const float* pb = AB + ((size_t)(j0 + 16 * c + mn) * N_ + ka) * LD + CZ_ + d;
        b.x = pb[0]; b.y = pb[LD];
      } else {
        const float* pb = AB + ((size_t)ka * N_ + j0 + 16 * c + mn) * LD + CZ_ + d;
        b.x = pb[0]; b.y = pb[(size_t)N_ * LD];
      }
      acc[c] = wmma4(a, b, acc[c]);
    }
  }
#pragma unroll
  for (int c = 0; c < 4; ++c) {
#pragma unroll
    for (int r = 0; r < 8; ++r) {
      int i = i0 + half * 8 + r;
      O[((size_t)i * N_ + j0 + 16 * c + mn) * CZ_ + d] = acc[c][r];
    }
  }
}

// ---------------------------------------------------------------------------
// Unified attention: per (batch i, head h, 16-query tile):
//   S = Q K^T * scale + bias[h,q,k];  P = softmax_k(S);  O = P V
// Q/K/V/O layout: [i, pos, h, 32] (row stride = nH*32).
// bias layout: bias[(h*256 + q)*256 + k] (broadcast over i).
// One wave per q-tile; scores staged through LDS (16x256 fp32 per wave).
// ---------------------------------------------------------------------------
__global__ void attn_kernel(const float* __restrict__ Q, const float* __restrict__ K,
                            const float* __restrict__ V, const float* __restrict__ bias,
                            float* __restrict__ O, int nH, float scale) {
  __shared__ float smem[4][16 * N_];
  int i = blockIdx.z;
  int h = blockIdx.y;
  int q0 = (blockIdx.x * blockDim.y + threadIdx.y) * 16;
  int ld = nH * 32;
  const float* Qb = Q + ((size_t)i * N_ * nH + h) * 32;
  const float* Kb = K + ((size_t)i * N_ * nH + h) * 32;
  const float* Vb = V + ((size_t)i * N_ * nH + h) * 32;
  float* P = smem[threadIdx.y];
  int lane = threadIdx.x & 31;
  int n = lane & 15, half = lane >> 4;

  // scores -> LDS
  for (int kt = 0; kt < N_; kt += 16) {
    v8f acc = {};
#pragma unroll
    for (int k0 = 0; k0 < 32; k0 += 4) {
      v2f a = load_a_tile(Qb, ld, q0, k0);
      int dd = k0 + half * 2;  // B operand: [dd][n] = Kb[(kt+n)*ld + dd]
      const float* pb = Kb + (size_t)(kt + n) * ld + dd;
      v2f b; b.x = pb[0]; b.y = pb[1];
      acc = wmma4(a, b, acc);
    }
#pragma unroll
    for (int r = 0; r < 8; ++r) {
      int q = half * 8 + r;
      int k = kt + n;
      P[q * N_ + k] = acc[r] * scale + bias[((size_t)h * N_ + (q0 + q)) * N_ + k];
    }
  }

  // softmax over k (each lane owns half of one row; combine halves via shfl)
  int rr = lane & 15;
  int kh = (lane >> 4) * (N_ / 2);
  float mx = -3.4e38f;
  for (int k = 0; k < N_ / 2; ++k) mx = fmaxf(mx, P[rr * N_ + kh + k]);
  mx = fmaxf(mx, __shfl_xor(mx, 16));
  float sum = 0.f;
  for (int k = 0; k < N_ / 2; ++k) {
    float e = __expf(P[rr * N_ + kh + k] - mx);
    P[rr * N_ + kh + k] = e;
    sum += e;
  }
  sum += __shfl_xor(sum, 16);
  float inv = 1.f / sum;
  for (int k = 0; k < N_ / 2; ++k) P[rr * N_ + kh + k] *= inv;

  // O = P @ V  (16 x 32), P read back from LDS in A-operand layout
  float* Ob = O + ((size_t)i * N_ * nH + h) * 32;
#pragma unroll
  for (int d0 = 0; d0 < 32; d0 += 16) {
    v8f acc = {};
    for (int k0 = 0; k0 < N_; k0 += 4) {
      v2f a = load_a_tile(P, N_, 0, k0);
      v2f b = load_b_tile(Vb, ld, k0, d0);
      acc = wmma4(a, b, acc);
    }
#pragma unroll
    for (int r = 0; r < 8; ++r) {
      int q = q0 + half * 8 + r;
      Ob[(size_t)q * ld + d0 + n] = acc[r];
    }
  }
}

// ---------------------------------------------------------------------------
// LayerNorm over last dim D (wave per row, D/32 elems per lane).
// TRANSP: source row (i*256+q) is read from (q*256+i)   (triangle "ending").
// ---------------------------------------------------------------------------
template <int D, bool TRANSP>
__global__ void ln_kernel(const float* __restrict__ x, const float* __restrict__ w,
                          const float* __restrict__ b, float* __restrict__ y) {
  int lane = threadIdx.x;
  int row = blockIdx.x * blockDim.y + threadIdx.y;
  int src = TRANSP ? (((row & 255) << 8) | (row >> 8)) : row;
  const float* xp = x + (size_t)src * D;
  const int V = D / 32;
  float v[V];
  float s = 0.f;
#pragma unroll
  for (int i = 0; i < V; ++i) { v[i] = xp[lane + i * 32]; s += v[i]; }
#pragma unroll
  for (int m = 16; m >= 1; m >>= 1) s += __shfl_xor(s, m);
  float mean = s / D;
  float q = 0.f;
#pragma unroll
  for (int i = 0; i < V; ++i) { float d = v[i] - mean; q += d * d; }
#pragma unroll
  for (int m = 16; m >= 1; m >>= 1) q += __shfl_xor(q, m);
  float invs = rsqrtf(q / D + EPS_);
  float* yp = y + (size_t)row * D;
#pragma unroll
  for (int i = 0; i < V; ++i) {
    int c = lane + i * 32;
    yp[c] = (v[i] - mean) * invs * w[c] + b[c];
  }
}

// Small projection (K -> Hn, Hn in {4,12}); output column-major: Y[h*rows + r]
__global__ void proj_small_kernel(const float* __restrict__ X, const float* __restrict__ W,
                                  float* __restrict__ Y, int rows, int Kd, int Hn) {
  int idx = blockIdx.x * blockDim.x + threadIdx.x;
  if (idx >= rows * Hn) return;
  int r = idx / Hn, h = idx - r * Hn;
  const float* xp = X + (size_t)r * Kd;
  float s = 0.f;
  for (int k = 0; k < Kd; ++k) s += xp[k] * W[k * Hn + h];
  Y[(size_t)h * rows + r] = s;
}

__global__ void mul_inplace_kernel(float* __restrict__ a, const float* __restrict__ g, int n) {
  int i = blockIdx.x * blockDim.x + threadIdx.x;
  if (i < n) a[i] *= g[i];
}

// ---------------------------------------------------------------------------
// Host-side launch wrappers
// ---------------------------------------------------------------------------
template <int EPI, bool TR>
static void run_gemm(const float* A, const float* B, const float* bias, float* C,
                     int M, int Nn, int K, int lda, int ldb, int ldc, hipStream_t st) {
  dim3 blk(32, 4), grd(Nn / 64, M / 64);
  gemm_kernel<EPI, TR><<<grd, blk, 0, st>>>(A, B, bias, C, K, lda, ldb, ldc);
}

template <int GM, bool ACC, bool SAME_A>
static void run_dual(const float* A1, const float* B1, const float* A2, const float* B2,
                     float* C, int M, int Nn, int K, int lda, int ldb, int ldc,
                     hipStream_t st) {
  dim3 blk(32, 4), grd(Nn / 32, M / 64);
  gemm_dual_kernel<GM, ACC, SAME_A><<<grd, blk, 0, st>>>(A1, B1, A2, B2, C, K, lda, ldb, ldc);
}

template <int D, bool TR>
static void run_ln(const float* x, const float* w, const float* b, float* y, int rows,
                   hipStream_t st) {
  dim3 blk(32, 8), grd(rows / 8);
  ln_kernel<D, TR><<<grd, blk, 0, st>>>(x, w, b, y);
}

template <bool E>
static void run_einsum(const float* AB, float* O, hipStream_t st) {
  dim3 blk(32, 4), grd(1, N_ / 16, CZ_);
  trimul_einsum_kernel<E><<<grd, blk, 0, st>>>(AB, O);
}

static void run_attn(const float* Q, const float* K, const float* V, const float* bias,
                     float* O, int nI, int nH, float scale, hipStream_t st) {
  dim3 blk(32, 4), grd(N_ / 64, nH, nI);
  attn_kernel<<<grd, blk, 0, st>>>(Q, K, V, bias, O, nH, scale);
}

static void run_proj(const float* X, const float* W, float* Y, int rows, int Kd, int Hn,
                     hipStream_t st) {
  int total = rows * Hn;
  proj_small_kernel<<<(total + 255) / 256, 256, 0, st>>>(X, W, Y, rows, Kd, Hn);
}

static void run_mul(float* a, const float* g, int n, hipStream_t st) {
  mul_inplace_kernel<<<(n + 255) / 256, 256, 0, st>>>(a, g, n);
}

// ---------------------------------------------------------------------------
// kernel_launch: full Pairformer layer
// Inputs: 0:s 1:z, tmo:2-9, tmi:10-17, tas:18-25, tae:26-33, tz:34-38,
//         apb:39-49, ts:50-54
// ---------------------------------------------------------------------------
extern "C" void kernel_launch(void* const* d_in, const int* in_sizes, int n_in,
                              void* d_out, int out_size, void* d_ws, size_t ws_size,
                              hipStream_t stream) {
  (void)in_sizes; (void)n_in; (void)out_size; (void)ws_size;
  auto in = [&](int i) { return (const float*)d_in[i]; };

  float* S = (float*)d_out;                 // (256, 384)
  float* Z = S + (size_t)N_ * CS_;          // (256, 256, 128)

  float* ws = (float*)d_ws;
  float* XN  = ws;                                          // ZS*128
  float* BIG = ws + (size_t)ZS_ * CZ_;                      // ZS*512
  float* TB  = BIG + (size_t)ZS_ * 4 * CZ_;                 // 12*ZS (bias / zb)
  float* SN  = TB + (size_t)12 * ZS_;                       // 256*384 each:
  float* QS  = SN + (size_t)N_ * CS_;
  float* KS  = QS + (size_t)N_ * CS_;
  float* VS  = KS + (size_t)N_ * CS_;
  float* GS  = VS + (size_t)N_ * CS_;
  float* OS  = GS + (size_t)N_ * CS_;

  const float invsqrt32 = 0.17677669529663689f;

  hipMemcpyAsync(S, in(0), (size_t)N_ * CS_ * sizeof(float),
                 hipMemcpyDeviceToDevice, stream);
  hipMemcpyAsync(Z, in(1), (size_t)ZS_ * CZ_ * sizeof(float),
                 hipMemcpyDeviceToDevice, stream);

  // ---- tri_mul (tmo: ending=false, tmi: ending=true) ----
  auto tri_mul = [&](int base, bool ending) {
    run_ln<CZ_, false>(Z, in(base + 0), in(base + 1), XN, ZS_, stream);
    // AB = (XN@pin) * sigmoid(XN@gin)  -> (ZS, 256)
    run_dual<GM_SIG, false, true>(XN, in(base + 2), XN, in(base + 3), BIG,
                                  ZS_, 2 * CZ_, CZ_, CZ_, 2 * CZ_, 2 * CZ_, stream);
    float* O  = BIG + (size_t)ZS_ * 2 * CZ_;
    float* ON = BIG + (size_t)ZS_ * 3 * CZ_;
    if (ending) run_einsum<true>(BIG, O, stream);
    else        run_einsum<false>(BIG, O, stream);
    run_ln<CZ_, false>(O, in(base + 4), in(base + 5), ON, ZS_, stream);
    // Z += (LN(O)@pout) * sigmoid(XN@gout)
    run_dual<GM_SIG, true, false>(ON, in(base + 6), XN, in(base + 7), Z,
                                  ZS_, CZ_, CZ_, CZ_, CZ_, CZ_, stream);
  };
  tri_mul(2, false);
  tri_mul(10, true);

  // ---- tri_att (tas: ending=false, tae: ending=true) ----
  auto tri_att = [&](int base, bool ending) {
    if (ending) run_ln<CZ_, true>(Z, in(base + 0), in(base + 1), XN, ZS_, stream);
    else        run_ln<CZ_, false>(Z, in(base + 0), in(base + 1), XN, ZS_, stream);
    run_proj(XN, in(base + 2), TB, ZS_, CZ_, 4, stream);  // triangle bias
    float* Qb = BIG;
    float* Kb = BIG + (size_t)ZS_ * CZ_;
    float* Vb = BIG + (size_t)ZS_ * 2 * CZ_;
    float* Ob = BIG + (size_t)ZS_ * 3 * CZ_;
    run_gemm<EPI_STORE, false>(XN, in(base + 3), nullptr, Qb, ZS_, CZ_, CZ_,
                               CZ_, CZ_, CZ_, stream);
    run_gemm<EPI_STORE, false>(XN, in(base + 4), nullptr, Kb, ZS_, CZ_, CZ_,
                               CZ_, CZ_, CZ_, stream);
    run_gemm<EPI_STORE, false>(XN, in(base + 5), nullptr, Vb, ZS_, CZ_, CZ_,
                               CZ_, CZ_, CZ_, stream);
    run_attn(Qb, Kb, Vb, TB, Ob, N_, 4, invsqrt32, stream);
    // gate = sigmoid(XN@g) (reuse Qb), O *= gate, Z += O@o (transposed if ending)
    run_gemm<EPI_SIG, false>(XN, in(base + 6), nullptr, Qb, ZS_, CZ_, CZ_,
                             CZ_, CZ_, CZ_, stream);
    run_mul(Ob, Qb, ZS_ * CZ_, stream);
    if (ending)
      run_gemm<EPI_ACC, true>(Ob, in(base + 7), nullptr, Z, ZS_, CZ_, CZ_,
                              CZ_, CZ_, CZ_, stream);
    else
      run_gemm<EPI_ACC, false>(Ob, in(base + 7), nullptr, Z, ZS_, CZ_, CZ_,
                               CZ_, CZ_, CZ_, stream);
  };
  tri_att(18, false);
  tri_att(26, true);

  // ---- pair transition (tz) : Z += (silu(XN@f1)*(XN@f2)) @ f3 ----
  run_ln<CZ_, false>(Z, in(34), in(35), XN, ZS_, stream);
  run_dual<GM_SILU, false, true>(XN, in(36), XN, in(37), BIG,
                                 ZS_, 4 * CZ_, CZ_, CZ_, 4 * CZ_, 4 * CZ_, stream);
  run_gemm<EPI_ACC, false>(BIG, in(38), nullptr, Z, ZS_, CZ_, 4 * CZ_,
                           4 * CZ_, CZ_, CZ_, stream);

  // ---- attention-with-pair-bias (apb) ----
  run_ln<CS_, false>(S, in(39), in(40), SN, N_, stream);
  run_gemm<EPI_BIAS, false>(SN, in(41), in(42), QS, N_, CS_, CS_, CS_, CS_, CS_, stream);
  run_gemm<EPI_STORE, false>(SN, in(43), nullptr, KS, N_, CS_, CS_, CS_, CS_, CS_, stream);
  run_gemm<EPI_STORE, false>(SN, in(44), nullptr, VS, N_, CS_, CS_, CS_, CS_, CS_, stream);
  run_gemm<EPI_SIG, false>(SN, in(45), nullptr, GS, N_, CS_, CS_, CS_, CS_, CS_, stream);
  run_ln<CZ_, false>(Z, in(46), in(47), XN, ZS_, stream);
  run_proj(XN, in(48), TB, ZS_, CZ_, 12, stream);          // zb bias
  run_attn(QS, KS, VS, TB, OS, 1, 12, invsqrt32, stream);
  run_mul(OS, GS, N_ * CS_, stream);
  run_gemm<EPI_ACC, false>(OS, in(49), nullptr, S, N_, CS_, CS_, CS_, CS_, CS_, stream);

  // ---- single transition (ts) : S += (silu(SN@f1)*(SN@f2)) @ f3 ----
  run_ln<CS_, false>(S, in(50), in(51), SN, N_, stream);
  run_dual<GM_SILU, false, true>(SN, in(52), SN, in(53), BIG,
                                 N_, 4 * CS_, CS_, CS_, 4 * CS_, 4 * CS_, stream);
  run_gemm<EPI_ACC, false>(BIG, in(54), nullptr, S, N_, CS_, 4 * CS_,
                           4 * CS_, CS_, CS_, stream);
}